// LinearEncoder_84267258348157
// MI455X (gfx1250) — compile-verified
//
#include <hip/hip_runtime.h>
#include <hip/hip_bf16.h>

#define N_NODES 50000
#define N_EDGES 800000
#define D_FEAT  64

typedef __attribute__((ext_vector_type(2))) float v2f;
typedef __attribute__((ext_vector_type(8))) float v8f;

// ---------------------------------------------------------------------------
// Kernel 1: deg[i] = 1.0 (self-loop)
// ---------------------------------------------------------------------------
__global__ void gcn_init_deg(float* __restrict__ deg) {
    int i = blockIdx.x * blockDim.x + threadIdx.x;
    if (i < N_NODES) deg[i] = 1.0f;
}

// ---------------------------------------------------------------------------
// Kernel 2: deg[dst[e]] += 1  (atomic, L2-resident 200KB array)
// ---------------------------------------------------------------------------
__global__ void gcn_count_deg(const int* __restrict__ dst, float* __restrict__ deg) {
    int e = blockIdx.x * blockDim.x + threadIdx.x;
    if (e < N_EDGES) atomicAdd(&deg[dst[e]], 1.0f);
}

// ---------------------------------------------------------------------------
// Kernel 3: dis[i] = rsqrt(deg[i])   (deg >= 1 always, no zero guard needed)
// ---------------------------------------------------------------------------
__global__ void gcn_deg_inv_sqrt(const float* __restrict__ deg, float* __restrict__ dis) {
    int i = blockIdx.x * blockDim.x + threadIdx.x;
    if (i < N_NODES) dis[i] = rsqrtf(deg[i]);
}

// ---------------------------------------------------------------------------
// Kernel 4: XW = X @ W via V_WMMA_F32_16X16X4_F32.
// 3125 blocks x 128 threads (4 waves). Block = one 16-row tile of X.
// Wave w computes the 16x16 output tile at columns [16w, 16w+16).
// X tile (16x64, 4KB) and all of W (64x64, 16KB) staged in LDS.
// 50000 % 16 == 0 -> no tail handling.
// ---------------------------------------------------------------------------
__global__ __launch_bounds__(128) void gcn_gemm_xw(const float* __restrict__ X,
                                                   const float* __restrict__ W,
                                                   float* __restrict__ XW) {
    __shared__ float Xs[16 * D_FEAT];   // 4 KB
    __shared__ float Ws[D_FEAT * D_FEAT]; // 16 KB

    const int tid  = threadIdx.x;
    const int lane = tid & 31;
    const int wave = tid >> 5;          // column tile 0..3
    const int rt   = blockIdx.x;        // row tile 0..3124

    // Stage X tile: 1024 floats / 128 threads = 8 each (coalesced b32 loads)
    const float* xsrc = X + (size_t)rt * 16 * D_FEAT;
    #pragma unroll
    for (int i = 0; i < 8; ++i) Xs[tid + i * 128] = xsrc[tid + i * 128];
    // Stage W: 4096 floats / 128 threads = 32 each
    #pragma unroll
    for (int i = 0; i < 32; ++i) Ws[tid + i * 128] = W[tid + i * 128];
    __syncthreads();

    // ISA 7.12.2 f32 fragment layouts (wave32):
    //  A 16x4 : lanes 0-15 hold row M=lane,   v0=K0, v1=K1
    //           lanes 16-31 hold row M=lane-16, v0=K2, v1=K3
    //  B 4x16 : lanes 0-15 hold col N=lane,   v0=K0 row, v1=K1 row
    //           lanes 16-31 hold col N=lane-16, v0=K2 row, v1=K3 row
    const int m    = lane & 15;
    const int koff = (lane >> 4) * 2;   // 0 for low half, 2 for high half
    const int ncol = wave * 16 + (lane & 15);

    v8f acc = {};
    #pragma unroll
    for (int k0 = 0; k0 < D_FEAT; k0 += 4) {
        const int k = k0 + koff;
        v2f a, b;
        a.x = Xs[m * D_FEAT + k];
        a.y = Xs[m * D_FEAT + k + 1];
        b.x = Ws[k * D_FEAT + ncol];
        b.y = Ws[(k + 1) * D_FEAT + ncol];
        // (neg_a, A, neg_b, B, c_mod, C, reuse_a, reuse_b)
        acc = __builtin_amdgcn_wmma_f32_16x16x4_f32(
            false, a, false, b, (short)0, acc, false, false);
    }

    // C/D layout: VGPR v -> M = v (lanes 0-15) / v+8 (lanes 16-31), N = lane&15
    const int rbase = rt * 16 + ((lane >> 4) << 3);
    #pragma unroll
    for (int v = 0; v < 8; ++v) {
        XW[(size_t)(rbase + v) * D_FEAT + ncol] = acc[v];
    }
}

// ---------------------------------------------------------------------------
// Kernel 5: out[i,f] = xw[i,f] * dis[i]^2 + b[f]    (self-loop term + bias,
// also serves as the deterministic re-initialization of d_out)
// ---------------------------------------------------------------------------
__global__ void gcn_init_out(const float* __restrict__ XW,
                             const float* __restrict__ dis,
                             const float* __restrict__ bias,
                             float* __restrict__ out) {
    int g = blockIdx.x * blockDim.x + threadIdx.x;
    if (g < N_NODES * D_FEAT) {
        int node = g >> 6;
        int f    = g & 63;
        float s  = dis[node];
        out[g] = XW[g] * (s * s) + bias[f];
    }
}

// ---------------------------------------------------------------------------
// Kernel 6: per edge e, out[dst[e],:] += xw[src[e],:] * dis[src]*dis[dst]
// 64 threads per edge -> coalesced 256B row gather + f32 atomics (L2 RMW).
// ---------------------------------------------------------------------------
__global__ void gcn_scatter(const int* __restrict__ src,
                            const int* __restrict__ dst,
                            const float* __restrict__ XW,
                            const float* __restrict__ dis,
                            float* __restrict__ out) {
    long long g = (long long)blockIdx.x * blockDim.x + threadIdx.x;
    int e = (int)(g >> 6);
    int f = (int)(g & 63);
    if (e < N_EDGES) {
        int s = src[e];
        int d = dst[e];
        float norm = dis[s] * dis[d];
        atomicAdd(&out[(size_t)d * D_FEAT + f], XW[(size_t)s * D_FEAT + f] * norm);
    }
}

// ---------------------------------------------------------------------------
extern "C" void kernel_launch(void* const* d_in, const int* in_sizes, int n_in,
                              void* d_out, int out_size, void* d_ws, size_t ws_size,
                              hipStream_t stream) {
    const float* X     = (const float*)d_in[0];
    const int*   edges = (const int*)d_in[1];     // [2, N_EDGES]: row0=src, row1=dst
    const float* W     = (const float*)d_in[2];
    const float* bias  = (const float*)d_in[3];
    float*       out   = (float*)d_out;

    const int* src = edges;
    const int* dst = edges + N_EDGES;

    // Workspace layout
    float* xw  = (float*)d_ws;               // N_NODES * 64 floats (12.8 MB)
    float* deg = xw + (size_t)N_NODES * D_FEAT;  // N_NODES floats
    float* dis = deg + N_NODES;                  // N_NODES floats

    const int T = 256;

    gcn_init_deg   <<<(N_NODES + T - 1) / T, T, 0, stream>>>(deg);
    gcn_count_deg  <<<(N_EDGES + T - 1) / T, T, 0, stream>>>(dst, deg);
    gcn_deg_inv_sqrt<<<(N_NODES + T - 1) / T, T, 0, stream>>>(deg, dis);

    gcn_gemm_xw    <<<N_NODES / 16, 128, 0, stream>>>(X, W, xw);

    gcn_init_out   <<<(N_NODES * D_FEAT + T - 1) / T, T, 0, stream>>>(xw, dis, bias, out);

    long long scat_threads = (long long)N_EDGES * D_FEAT;
    gcn_scatter    <<<(unsigned)((scat_threads + T - 1) / T), T, 0, stream>>>(src, dst, xw, dis, out);
}